// DecodePredictions_41403484733600
// MI455X (gfx1250) — compile-verified
//
#include <hip/hip_runtime.h>
#include <hip/hip_bf16.h>

// ---------------------------------------------------------------------------
// RetinaNet decode + per-class NMS for MI455X (gfx1250, wave32).
// predictions: [8, 76725, 84] f32  ->  boxes[8,50,4], scores[8,50],
// classes[8,50], valid[8]  (2408 floats concatenated in d_out).
// ---------------------------------------------------------------------------

#define BATCH   8
#define NANCH   76725
#define NCLS    80
#define NCH     84
#define KC      256          // candidates fed to greedy NMS (== K_CAND)
#define CANDCAP 2048         // per-class candidate buffer (power of 2 for bitonic)
#define NBINS   4096         // 12-bit radix histogram
#define MAXDET  50
// sigmoid(x) > 0.05  <=>  x > ln(0.05/0.95)
#define LOGIT_THR (-2.9444389791664403f)

typedef float v2f __attribute__((ext_vector_type(2)));
typedef float v8f __attribute__((ext_vector_type(8)));

// -------- order-preserving float<->uint key ---------------------------------
__device__ __forceinline__ unsigned f2key(float f) {
  unsigned u = __float_as_uint(f);
  return (u & 0x80000000u) ? ~u : (u | 0x80000000u);
}
__device__ __forceinline__ float key2f(unsigned k) {
  unsigned u = (k & 0x80000000u) ? (k ^ 0x80000000u) : ~k;
  return __uint_as_float(u);
}

// -------- RetinaNet anchor for flat index a (H=W=640) -----------------------
__device__ __forceinline__ void anchor_of(int a, float& cx, float& cy,
                                          float& aw, float& ah) {
  int base, fw; float stride;
  if      (a < 57600) { base = 0;     fw = 80; stride = 8.f;   }
  else if (a < 72000) { base = 57600; fw = 40; stride = 16.f;  }
  else if (a < 75600) { base = 72000; fw = 20; stride = 32.f;  }
  else if (a < 76500) { base = 75600; fw = 10; stride = 64.f;  }
  else                { base = 76500; fw = 5;  stride = 128.f; }
  int local = a - base;
  int k    = local % 9;
  int cell = local / 9;
  int x = cell % fw, y = cell / fw;
  cx = (x + 0.5f) * stride;
  cy = (y + 0.5f) * stride;
  const float sr[3] = {0.70710678118f, 1.0f, 1.41421356237f};   // sqrt(ratio)
  const float sc[3] = {1.0f, 1.25992104989f, 1.58740105197f};   // 2^(k/3)
  float s4 = 4.0f * stride;                                     // sqrt(area)
  int ri = k / 3, si = k % 3;
  aw = sc[si] * s4 * sr[ri];   // width  = s * area/h
  ah = sc[si] * s4 / sr[ri];   // height = s * sqrt(area/ratio)
}

// -------- descending bitonic sort of u64 keys in LDS ------------------------
__device__ __forceinline__ void bitonic_desc(unsigned long long* s, int N,
                                             int tid, int nthr) {
  for (int k = 2; k <= N; k <<= 1) {
    for (int j = k >> 1; j > 0; j >>= 1) {
      for (int i = tid; i < N; i += nthr) {
        int ixj = i ^ j;
        if (ixj > i) {
          unsigned long long a = s[i], b = s[ixj];
          bool up = ((i & k) == 0);               // descending half
          if (up ? (a < b) : (a > b)) { s[i] = b; s[ixj] = a; }
        }
      }
      __syncthreads();
    }
  }
}

// ===========================================================================
// K0: zero histogram / counters region of workspace
// ===========================================================================
__global__ __launch_bounds__(256) void zero_kernel(unsigned* p, long long n) {
  long long i = (long long)blockIdx.x * blockDim.x + threadIdx.x;
  long long step = (long long)gridDim.x * blockDim.x;
  for (; i < n; i += step) p[i] = 0u;
}

// ===========================================================================
// K1: decode anchor boxes -> corner boxes  [B*A, 4]
// ===========================================================================
__global__ __launch_bounds__(256) void decode_kernel(
    const float* __restrict__ pred, float* __restrict__ boxes) {
  int t = blockIdx.x * blockDim.x + threadIdx.x;
  if (t >= BATCH * NANCH) return;
  int a = t % NANCH;
  const float* p = pred + (size_t)t * NCH;
  float cx, cy, aw, ah;
  anchor_of(a, cx, cy, aw, ah);
  float bx = p[0] * 0.1f * aw + cx;
  float by = p[1] * 0.1f * ah + cy;
  float bw = __expf(p[2] * 0.2f) * aw;
  float bh = __expf(p[3] * 0.2f) * ah;
  float* o = boxes + (size_t)t * 4;
  o[0] = bx - 0.5f * bw; o[1] = by - 0.5f * bh;
  o[2] = bx + 0.5f * bw; o[3] = by + 0.5f * bh;
}

// ===========================================================================
// K2a: coalesced radix histogram of logit keys per (b, class)
// ===========================================================================
__global__ __launch_bounds__(256) void hist_kernel(
    const float* __restrict__ pred, unsigned* __restrict__ gHist) {
  long long t = (long long)blockIdx.x * blockDim.x + threadIdx.x;
  if (t >= (long long)BATCH * NANCH * NCLS) return;
  int c = (int)(t % NCLS);
  long long r = t / NCLS;
  int a = (int)(r % NANCH);
  int b = (int)(r / NANCH);
  float logit = pred[((size_t)b * NANCH + a) * NCH + 4 + c];
  if (logit > LOGIT_THR) {
    unsigned key = f2key(logit);
    atomicAdd(&gHist[(((size_t)b * NCLS + c) << 12) + (key >> 20)], 1u);
  }
}

// ===========================================================================
// K2b: per-(b,c) find radix bin T such that count(bin >= T) >= 256 (top-K)
// ===========================================================================
__global__ __launch_bounds__(256) void thresh_kernel(
    const unsigned* __restrict__ gHist, int* __restrict__ Tbin) {
  __shared__ unsigned h[NBINS];
  int bc = blockIdx.x;
  for (int i = threadIdx.x; i < NBINS; i += 256)
    h[i] = gHist[((size_t)bc << 12) + i];
  __syncthreads();
  if (threadIdx.x == 0) {
    int sum = 0, T = 0;
    for (int bin = NBINS - 1; bin >= 0; --bin) {
      sum += (int)h[bin];
      if (sum >= KC) { T = bin; break; }
    }
    Tbin[bc] = T;
  }
}

// ===========================================================================
// K2c: coalesced collect of top-bin candidates into per-class buffers
// ===========================================================================
__global__ __launch_bounds__(256) void collect_kernel(
    const float* __restrict__ pred, const int* __restrict__ Tbin,
    int* __restrict__ candCount, unsigned long long* __restrict__ candBuf) {
  long long t = (long long)blockIdx.x * blockDim.x + threadIdx.x;
  if (t >= (long long)BATCH * NANCH * NCLS) return;
  int c = (int)(t % NCLS);
  long long r = t / NCLS;
  int a = (int)(r % NANCH);
  int b = (int)(r / NANCH);
  float logit = pred[((size_t)b * NANCH + a) * NCH + 4 + c];
  if (logit > LOGIT_THR) {
    unsigned key = f2key(logit);
    int bc = b * NCLS + c;
    if ((int)(key >> 20) >= Tbin[bc]) {
      int pos = atomicAdd(&candCount[bc], 1);
      if (pos < CANDCAP)
        candBuf[(size_t)bc * CANDCAP + pos] =
            ((unsigned long long)key << 32) | (0xFFFFFFFFu - (unsigned)a);
    }
  }
}

// ===========================================================================
// K3: per-(b,c) sort candidates + greedy NMS.
//     256x256 suppression matrix from 16x16 tiles: the union term
//     (area_i + area_j) is V_WMMA_F32_16X16X4_F32 of [area_a,1]x[[1],[area_b]];
//     VALU finishes the intersection and division-free IoU>0.5 compare.
//     Each wave owns two tile-ROWS and sweeps all 16 tile-columns, so
//     suppression words are accumulated in registers (ballot per D-row) and
//     published with plain ds_store — no LDS atomics, no mask zero-init.
// ===========================================================================
__global__ __launch_bounds__(256) void nms_kernel(
    const unsigned long long* __restrict__ candBuf,
    const int* __restrict__ candCount,
    const float* __restrict__ boxes,
    float* __restrict__ selBoxes, float* __restrict__ selScores) {
  __shared__ unsigned long long s[CANDCAP];   // 16 KB sort buffer
  __shared__ float bx[KC][4];                 // candidate corner boxes
  __shared__ float area[KC];
  __shared__ float sscore[KC];                // sigmoid score (-1 = invalid)
  __shared__ unsigned supp[KC * 8];           // 256x256 suppression bitmask
  __shared__ int keepIdx[MAXDET];
  __shared__ int keepCount;

  int bc = blockIdx.x, tid = threadIdx.x;
  int b = bc / NCLS;

  int n = candCount[bc];
  if (n > CANDCAP) n = CANDCAP;
  for (int i = tid; i < CANDCAP; i += 256)
    s[i] = (i < n) ? candBuf[(size_t)bc * CANDCAP + i] : 0ULL;
  __syncthreads();
  bitonic_desc(s, CANDCAP, tid, 256);         // score desc, anchor asc

  int m = (n < KC) ? n : KC;
  {   // unpack top-256: thread tid owns slot tid
    int slot = tid;
    if (slot < m) {
      unsigned long long kv = s[slot];
      unsigned key  = (unsigned)(kv >> 32);
      unsigned aidx = 0xFFFFFFFFu - (unsigned)(kv & 0xFFFFFFFFu);
      float logit = key2f(key);
      sscore[slot] = 1.0f / (1.0f + __expf(-logit));
      const float* bp = boxes + ((size_t)b * NANCH + aidx) * 4;
      float x1 = bp[0], y1 = bp[1], x2 = bp[2], y2 = bp[3];
      bx[slot][0] = x1; bx[slot][1] = y1; bx[slot][2] = x2; bx[slot][3] = y2;
      area[slot] = (x2 - x1) * (y2 - y1);
    } else {
      sscore[slot] = -1.0f;
      bx[slot][0] = bx[slot][1] = bx[slot][2] = bx[slot][3] = 0.f;
      area[slot] = 0.f;
    }
  }
  __syncthreads();

  // ---- WMMA suppression-mask build: wave owns tile-rows {wave, wave+8} ----
  int wave = tid >> 5, lane = tid & 31;
  int mofs = (lane < 16) ? 0 : 8;             // D-row half for this lane
  for (int tb = 0; tb < 2; ++tb) {
    int ti = wave + tb * 8;
    // A (16x4): col0 = area_a, col1 = 1, cols 2..3 = 0 (lanes 16-31: K=2,3)
    v2f Af;
    if (lane < 16) { Af[0] = area[ti * 16 + lane]; Af[1] = 1.0f; }
    else           { Af[0] = 0.f;                  Af[1] = 0.f;  }
    // this lane's 8 rows, hoisted to registers for the whole column sweep
    float rx1[8], ry1[8], rx2[8], ry2[8];
#pragma unroll
    for (int r = 0; r < 8; ++r) {
      int i = ti * 16 + r + mofs;
      rx1[r] = bx[i][0]; ry1[r] = bx[i][1];
      rx2[r] = bx[i][2]; ry2[r] = bx[i][3];
    }
    for (int tjp = 0; tjp < 8; ++tjp) {       // one 32-bit supp word per row
      unsigned acc[8];
#pragma unroll
      for (int r = 0; r < 8; ++r) acc[r] = 0u;
#pragma unroll
      for (int half = 0; half < 2; ++half) {
        int tj = tjp * 2 + half;
        // B (4x16): row0 = 1, row1 = area_b, rows 2..3 = 0
        v2f Bf;
        if (lane < 16) { Bf[0] = 1.0f; Bf[1] = area[tj * 16 + lane]; }
        else           { Bf[0] = 0.f;  Bf[1] = 0.f;                  }
        v8f Cz = {0.f, 0.f, 0.f, 0.f, 0.f, 0.f, 0.f, 0.f};
        // D[i][j] = area_a[i] + area_b[j]
        v8f D = __builtin_amdgcn_wmma_f32_16x16x4_f32(
            false, Af, false, Bf, (short)0, Cz, false, false);

        int j = tj * 16 + (lane & 15);
        float jx1 = bx[j][0], jy1 = bx[j][1], jx2 = bx[j][2], jy2 = bx[j][3];
#pragma unroll
        for (int r = 0; r < 8; ++r) {
          float w = fmaxf(fminf(rx2[r], jx2) - fmaxf(rx1[r], jx1), 0.f);
          float h = fmaxf(fminf(ry2[r], jy2) - fmaxf(ry1[r], jy1), 0.f);
          float inter = w * h;
          float uni   = D[r] - inter;          // WMMA outer-sum - inter
          // iou > 0.5  <=>  inter > 0.5*max(uni, 1e-8)   (division-free)
          bool sup = inter > 0.5f * fmaxf(uni, 1e-8f);
          unsigned bal = __builtin_amdgcn_ballot_w32(sup);
          // low 16 lanes -> row ti*16+r, high 16 lanes -> row ti*16+8+r
          unsigned mybits = (lane < 16) ? (bal & 0xFFFFu) : (bal >> 16);
          acc[r] |= mybits << (half * 16);
        }
      }
      // lanes 0 and 16 publish their half's 8 rows (wave-exclusive words)
      if ((lane & 15) == 0) {
#pragma unroll
        for (int r = 0; r < 8; ++r)
          supp[(ti * 16 + r + mofs) * 8 + tjp] = acc[r];
      }
    }
  }
  __syncthreads();

  // ---- greedy scan (matches the reference lax.scan semantics) -------------
  if (tid == 0) {
    unsigned kept[8] = {0, 0, 0, 0, 0, 0, 0, 0};
    int out = 0;
    for (int i = 0; i < KC && out < MAXDET; ++i) {
      if (sscore[i] <= 0.f) continue;              // top_s[i] > 0
      bool sup = false;
#pragma unroll
      for (int w = 0; w < 8; ++w)
        sup = sup || ((kept[w] & supp[i * 8 + w]) != 0u);
      if (!sup) {
        kept[i >> 5] |= 1u << (i & 31);
        keepIdx[out++] = i;
      }
    }
    keepCount = out;
  }
  __syncthreads();

  if (tid < MAXDET) {
    float* ob = selBoxes + ((size_t)bc * MAXDET + tid) * 4;
    if (tid < keepCount) {
      int i = keepIdx[tid];
      selScores[bc * MAXDET + tid] = sscore[i];
      ob[0] = bx[i][0]; ob[1] = bx[i][1]; ob[2] = bx[i][2]; ob[3] = bx[i][3];
    } else {
      selScores[bc * MAXDET + tid] = -1.0f;
      ob[0] = ob[1] = ob[2] = ob[3] = 0.f;
    }
  }
}

// ===========================================================================
// K4: per-batch combine -> final top-50 over 80*50 entries
// ===========================================================================
__global__ __launch_bounds__(256) void combine_kernel(
    const float* __restrict__ selBoxes, const float* __restrict__ selScores,
    float* __restrict__ out) {
  __shared__ unsigned long long s[4096];      // 32 KB
  int b = blockIdx.x, tid = threadIdx.x;
  const int NE = NCLS * MAXDET;               // 4000
  for (int i = tid; i < 4096; i += 256) {
    if (i < NE) {
      float sc = selScores[b * NE + i];
      s[i] = ((unsigned long long)f2key(sc) << 32) |
             (0xFFFFFFFFu - (unsigned)i);
    } else {
      s[i] = 0ULL;                            // below every real key
    }
  }
  __syncthreads();
  bitonic_desc(s, 4096, tid, 256);

  const int OB = 0;                            // boxes   [8,50,4]
  const int OS = BATCH * MAXDET * 4;           // scores  [8,50]   @1600
  const int OC = OS + BATCH * MAXDET;          // classes [8,50]   @2000
  const int OV = OC + BATCH * MAXDET;          // valid   [8]      @2400

  if (tid < MAXDET) {
    unsigned long long kv = s[tid];
    float sc = key2f((unsigned)(kv >> 32));
    unsigned slot = 0xFFFFFFFFu - (unsigned)(kv & 0xFFFFFFFFu);
    bool valid = sc > 0.f;
    float* ob = out + OB + ((size_t)b * MAXDET + tid) * 4;
    if (valid) {
      const float* bb = selBoxes + ((size_t)b * NE + slot) * 4;
      ob[0] = bb[0]; ob[1] = bb[1]; ob[2] = bb[2]; ob[3] = bb[3];
      out[OS + b * MAXDET + tid] = sc;
      out[OC + b * MAXDET + tid] = (float)(slot / MAXDET);
    } else {
      ob[0] = ob[1] = ob[2] = ob[3] = 0.f;
      out[OS + b * MAXDET + tid] = 0.f;
      out[OC + b * MAXDET + tid] = 0.f;
    }
  }
  if (tid == 0) {
    int cnt = 0;
    for (int i = 0; i < MAXDET; ++i) {
      float sc = key2f((unsigned)(s[i] >> 32));
      if (sc > 0.f) cnt++; else break;          // sorted desc -> can break
    }
    out[OV + b] = (float)cnt;
  }
}

// ===========================================================================
// host-side launcher
// ===========================================================================
static inline size_t align256(size_t x) { return (x + 255) & ~(size_t)255; }

extern "C" void kernel_launch(void* const* d_in, const int* in_sizes, int n_in,
                              void* d_out, int out_size, void* d_ws,
                              size_t ws_size, hipStream_t stream) {
  const float* pred = (const float*)d_in[1];   // d_in[0] = images (shape only)
  float* out = (float*)d_out;
  char* ws = (char*)d_ws;

  // workspace layout (~31 MB)
  size_t off = 0;
  size_t offBoxes = off; off = align256(off + (size_t)BATCH * NANCH * 4 * 4);
  size_t offHist  = off; off = align256(off + (size_t)BATCH * NCLS * NBINS * 4);
  size_t offTbin  = off; off = align256(off + (size_t)BATCH * NCLS * 4);
  size_t offCnt   = off; off = align256(off + (size_t)BATCH * NCLS * 4);
  size_t offCand  = off; off = align256(off + (size_t)BATCH * NCLS * CANDCAP * 8);
  size_t offSelB  = off; off = align256(off + (size_t)BATCH * NCLS * MAXDET * 4 * 4);
  size_t offSelS  = off; off = align256(off + (size_t)BATCH * NCLS * MAXDET * 4);
  (void)off; (void)ws_size; (void)in_sizes; (void)n_in; (void)out_size;

  float*              boxes     = (float*)(ws + offBoxes);
  unsigned*           gHist     = (unsigned*)(ws + offHist);
  int*                Tbin      = (int*)(ws + offTbin);
  int*                candCount = (int*)(ws + offCnt);
  unsigned long long* candBuf   = (unsigned long long*)(ws + offCand);
  float*              selBoxes  = (float*)(ws + offSelB);
  float*              selScores = (float*)(ws + offSelS);

  // K0: zero hist + Tbin + counters (contiguous region)
  long long zwords = (long long)(offCand - offHist) / 4;
  zero_kernel<<<4096, 256, 0, stream>>>(gHist, zwords);

  // K1: decode boxes
  int tot1 = BATCH * NANCH;
  decode_kernel<<<(tot1 + 255) / 256, 256, 0, stream>>>(pred, boxes);

  // K2a: histogram (coalesced full read of predictions' class channels)
  long long tot2 = (long long)BATCH * NANCH * NCLS;
  int blk2 = (int)((tot2 + 255) / 256);
  hist_kernel<<<blk2, 256, 0, stream>>>(pred, gHist);

  // K2b: per-class radix threshold
  thresh_kernel<<<BATCH * NCLS, 256, 0, stream>>>(gHist, Tbin);

  // K2c: collect candidates (second coalesced read)
  collect_kernel<<<blk2, 256, 0, stream>>>(pred, Tbin, candCount, candBuf);

  // K3: sort + WMMA-accelerated greedy NMS per (b, class)
  nms_kernel<<<BATCH * NCLS, 256, 0, stream>>>(candBuf, candCount, boxes,
                                               selBoxes, selScores);

  // K4: per-batch combine -> final detections
  combine_kernel<<<BATCH, 256, 0, stream>>>(selBoxes, selScores, out);
}